// PointPillarsScatter_446676599109
// MI455X (gfx1250) — compile-verified
//
#include <hip/hip_runtime.h>

// ---------------------------------------------------------------------------
// PointPillars scatter for MI455X (gfx1250).
//
// Phase 1: zero the 256 MB canvas using the CDNA5 async LDS->memory store
//          path (global_store_async_from_lds_b128, tracked by ASYNCcnt).
//          A 16B zero pattern is staged in LDS once per lane, then streamed
//          to global memory with fire-and-forget async stores.
// Phase 2: scatter. One thread per (pillar, channel). Wave32 => each wave
//          covers exactly one pillar, so coords are scalarized with
//          readfirstlane (SMEM loads). Feature reads are fully coalesced;
//          the 4B scattered writes use regular temporal hint so the 192 MB
//          L2 can merge them into full lines.
// ---------------------------------------------------------------------------

__global__ void __launch_bounds__(256)
pp_zero_canvas(float* __restrict__ out, unsigned int n4, unsigned int n)
{
    // 16 bytes of zeros per lane, staged in LDS (this __shared__ array is the
    // only LDS object, so it starts at LDS offset 0 of the workgroup).
    __shared__ __align__(16) float zbuf[256 * 4];
    float4* zslot = reinterpret_cast<float4*>(&zbuf[threadIdx.x * 4]);
    *zslot = make_float4(0.f, 0.f, 0.f, 0.f);     // ds_store_b128
    __syncthreads();                               // waits DScnt, barrier

    const unsigned int lds_off = threadIdx.x * 16u;
    const unsigned int stride  = gridDim.x * blockDim.x;
    float4* __restrict__ out4  = reinterpret_cast<float4*>(out);

    for (unsigned int i = blockIdx.x * blockDim.x + threadIdx.x; i < n4; i += stride) {
        float4* gp = out4 + i;
        // CDNA5 async store: copies 16 bytes from LDS offset lds_off to global address gp
        // Tracked with ASYNCcnt (hardware stalls issue before counter overflow).
        asm volatile("global_store_async_from_lds_b128 %0, %1, off"
                     :: "v"(gp), "v"(lds_off)
                     : "memory");
    }

    // Scalar tail for n % 4 (not hit for this problem size, kept for safety).
    for (unsigned int i = n4 * 4u + blockIdx.x * blockDim.x + threadIdx.x; i < n;
         i += stride) {
        out[i] = 0.f;
    }

    // Drain all outstanding async stores before the wave retires.
    asm volatile("s_wait_asynccnt 0" ::: "memory");
}

// Fast path: C is a compile-time power of two (C = 64 for PointPillars).
template <int CT>
__global__ void __launch_bounds__(256)
pp_scatter_fixedC(const float* __restrict__ feats,
                  const int*   __restrict__ coords,
                  const int*   __restrict__ nxp,
                  const int*   __restrict__ nyp,
                  float*       __restrict__ out,
                  int P)
{
    const int nx = nxp[0];            // uniform -> s_load
    const int ny = nyp[0];            // uniform -> s_load

    const int idx = blockIdx.x * blockDim.x + threadIdx.x;
    // 64 threads per pillar; wave32 => pillar id is uniform across the wave.
    const int p = __builtin_amdgcn_readfirstlane(idx >> 6);
    const int c = idx & (CT - 1);
    if (p >= P) return;

    // Uniform (scalar) coord loads: b, y, x shared by the whole wave.
    const int b = coords[p * 4 + 0];
    const int y = coords[p * 4 + 2];
    const int x = coords[p * 4 + 3];

    // Coalesced read: wave reads 128 contiguous bytes of this pillar's features.
    const float v = feats[(size_t)p * CT + c];

    // out[b][c][y][x] ; scattered 4B store, RT hint so L2 merges lines.
    const size_t o = ((size_t)(b * CT + c) * (size_t)ny + (size_t)y) * (size_t)nx
                   + (size_t)x;
    out[o] = v;
}

// Generic fallback for arbitrary C.
__global__ void __launch_bounds__(256)
pp_scatter_generic(const float* __restrict__ feats,
                   const int*   __restrict__ coords,
                   const int*   __restrict__ nxp,
                   const int*   __restrict__ nyp,
                   float*       __restrict__ out,
                   int P, int C)
{
    const int nx = nxp[0];
    const int ny = nyp[0];

    const int idx = blockIdx.x * blockDim.x + threadIdx.x;
    const int p = idx / C;
    const int c = idx - p * C;
    if (p >= P) return;

    const int b = coords[p * 4 + 0];
    const int y = coords[p * 4 + 2];
    const int x = coords[p * 4 + 3];

    const float v = feats[(size_t)p * C + c];
    const size_t o = ((size_t)(b * C + c) * (size_t)ny + (size_t)y) * (size_t)nx
                   + (size_t)x;
    out[o] = v;
}

extern "C" void kernel_launch(void* const* d_in, const int* in_sizes, int n_in,
                              void* d_out, int out_size, void* d_ws, size_t ws_size,
                              hipStream_t stream)
{
    (void)n_in; (void)d_ws; (void)ws_size;

    const float* feats  = (const float*)d_in[0];  // [P, C] fp32
    const int*   coords = (const int*)  d_in[1];  // [P, 4] int32 (b, z, y, x)
    // d_in[2] = batch_size (unused: b comes from coords, planes from out_size)
    const int*   nxp    = (const int*)  d_in[3];  // device scalar nx
    const int*   nyp    = (const int*)  d_in[4];  // device scalar ny
    float*       out    = (float*)      d_out;    // [B, C, ny, nx] fp32

    const int P = in_sizes[1] / 4;
    const int C = (P > 0) ? (in_sizes[0] / P) : 0;

    // ---- Phase 1: zero the canvas (async LDS->global streaming stores) ----
    const unsigned int n  = (unsigned int)out_size;
    const unsigned int n4 = n / 4u;
    pp_zero_canvas<<<dim3(4096), dim3(256), 0, stream>>>(out, n4, n);

    // ---- Phase 2: scatter ----
    if (P > 0 && C == 64) {
        const long long threads = (long long)P * 64;
        const int blocks = (int)((threads + 255) / 256);
        pp_scatter_fixedC<64><<<dim3(blocks), dim3(256), 0, stream>>>(
            feats, coords, nxp, nyp, out, P);
    } else if (P > 0 && C > 0) {
        const long long threads = (long long)P * C;
        const int blocks = (int)((threads + 255) / 256);
        pp_scatter_generic<<<dim3(blocks), dim3(256), 0, stream>>>(
            feats, coords, nxp, nyp, out, P, C);
    }
}